// IBPCDCNet_32315333935211
// MI455X (gfx1250) — compile-verified
//
#include <hip/hip_runtime.h>
#include <hip/hip_bf16.h>
#include <math.h>

// MI455X / gfx1250: wave32, WMMA (no MFMA). All channel-mixing GEMMs run on
// V_WMMA_F32_16X16X4_F32 at full fp32 reference precision (network is ~12 GFLOP
// total and L2-resident: latency-bound, no reason to drop precision).

typedef __attribute__((ext_vector_type(2))) float v2f;
typedef __attribute__((ext_vector_type(8))) float v8f;

// ---------------------------------------------------------------------------
// Generic batched GEMM on the WMMA pipe.
//   C[m,n] = sum_k A[m,k] * B[k,n]   (B optionally transposed: B[k,n]=Bm[n*ldb+k])
// Epilogue: +bias (per-row or per-col), *gamma+beta (folded BN, per-row),
//           +residual, ReLU. One wave (32 threads) computes one 16x16 C tile.
//
// Branch-free inner loop: out-of-range M rows / N cols use clamped addresses
// (their products only land in C rows/cols that are never stored, so no value
// masking is needed). Only the final K%4 chunk needs per-element masking of A.
// EXEC stays all-ones through every WMMA (ISA requirement).
// ---------------------------------------------------------------------------
__global__ __launch_bounds__(32)
void gemm_wmma_f32(const float* __restrict__ A, int lda, long long sA,
                   const float* __restrict__ Bm, int ldb, long long sB, int transB,
                   float* __restrict__ C, int ldc, long long sC,
                   const float* __restrict__ bias, long long sBias, int biasPerCol,
                   const float* __restrict__ gamma, const float* __restrict__ beta,
                   const float* __restrict__ resid, int ldres, long long sRes,
                   int M, int N, int K, int doRelu)
{
    const int z = blockIdx.z;
    A  += (long long)z * sA;
    Bm += (long long)z * sB;
    C  += (long long)z * sC;
    if (bias)  bias  += (long long)z * sBias;
    if (resid) resid += (long long)z * sRes;

    const int m0   = blockIdx.y << 4;
    const int n0   = blockIdx.x << 4;
    const int lane = threadIdx.x & 31;
    const int hl   = lane >> 4;      // half-wave: K pair select / M half on store
    const int l16  = lane & 15;
    const int am   = m0 + l16;       // A fragment row for this lane
    const int bn   = n0 + l16;       // B/C fragment column for this lane

    // Clamp addresses only (values need no masking for M/N overflow).
    const int amc = (am < M) ? am : (M - 1);
    const int bnc = (bn < N) ? bn : (N - 1);

    const float* Ap = A + (long long)amc * lda + (hl << 1);  // Ap[k0], Ap[k0+1]

    v8f acc = {};
    const int Kmain = K & ~3;

    if (transB) {
        const float* Bp = Bm + (long long)bnc * ldb + (hl << 1);
        for (int k0 = 0; k0 < Kmain; k0 += 4) {
            v2f af, bf;
            af[0] = Ap[k0];
            af[1] = Ap[k0 + 1];
            bf[0] = Bp[k0];
            bf[1] = Bp[k0 + 1];
            acc = __builtin_amdgcn_wmma_f32_16x16x4_f32(false, af, false, bf,
                                                        (short)0, acc, false, false);
        }
    } else {
        const float* Bq = Bm + (long long)(hl << 1) * ldb + bnc;  // Bq[k0*ldb], +ldb
        const long long ldb2 = (long long)ldb;
        for (int k0 = 0; k0 < Kmain; k0 += 4) {
            v2f af, bf;
            af[0] = Ap[k0];
            af[1] = Ap[k0 + 1];
            bf[0] = Bq[(long long)k0 * ldb2];
            bf[1] = Bq[(long long)k0 * ldb2 + ldb2];
            acc = __builtin_amdgcn_wmma_f32_16x16x4_f32(false, af, false, bf,
                                                        (short)0, acc, false, false);
        }
    }

    if (Kmain < K) {
        // one masked tail chunk: zeroed A elements kill invalid-k products,
        // B loads are merely address-clamped.
        const int ka  = Kmain + (hl << 1);
        const int k0c = (ka     < K) ? ka     : (K - 1);
        const int k1c = (ka + 1 < K) ? ka + 1 : (K - 1);
        v2f af, bf;
        af[0] = (ka     < K) ? A[(long long)amc * lda + k0c] : 0.0f;
        af[1] = (ka + 1 < K) ? A[(long long)amc * lda + k1c] : 0.0f;
        if (transB) {
            bf[0] = Bm[(long long)bnc * ldb + k0c];
            bf[1] = Bm[(long long)bnc * ldb + k1c];
        } else {
            bf[0] = Bm[(long long)k0c * ldb + bnc];
            bf[1] = Bm[(long long)k1c * ldb + bnc];
        }
        acc = __builtin_amdgcn_wmma_f32_16x16x4_f32(false, af, false, bf,
                                                    (short)0, acc, false, false);
    }

    if (bn >= N) return;
    const int mb = m0 + (hl << 3);   // lanes 0-15: rows m0..m0+7; lanes 16-31: +8
#pragma unroll
    for (int r = 0; r < 8; ++r) {
        const int m = mb + r;
        if (m >= M) continue;
        float v = acc[r];
        if (bias)  v += biasPerCol ? bias[bn] : bias[m];
        if (gamma) v = v * gamma[m] + (beta ? beta[m] : 0.0f);
        if (resid) v += resid[(long long)m * ldres + bn];
        if (doRelu) v = fmaxf(v, 0.0f);
        C[(long long)m * ldc + bn] = v;
    }
}

// ---------------------------------------------------------------------------
// (B,N,3) -> (B,3,N)
// ---------------------------------------------------------------------------
__global__ void transpose_in(const float* __restrict__ pcd, float* __restrict__ xyz,
                             int Bb, int N)
{
    int i = blockIdx.x * blockDim.x + threadIdx.x;
    int total = Bb * N * 3;
    if (i >= total) return;
    int c = i % 3;
    int n = (i / 3) % N;
    int b = i / (3 * N);
    xyz[((long long)b * 3 + c) * N + n] = pcd[i];
}

// ---------------------------------------------------------------------------
// Farthest point sampling: one block per batch, sequential argmax scan in LDS.
// Matches reference: dist init 1e10, first index 0, argmax tie -> lowest index.
// ---------------------------------------------------------------------------
__global__ __launch_bounds__(256)
void fps_kernel(const float* __restrict__ xyz, int* __restrict__ fidx,
                float* __restrict__ new_xyz, int N, int S)
{
    __shared__ float dist[2048];
    __shared__ float rv[256];
    __shared__ int   ri[256];
    __shared__ int   s_sel[256];
    __shared__ int   s_last;

    const int b  = blockIdx.x;
    const int t  = threadIdx.x;
    const int bd = blockDim.x;
    const float* X = xyz + (long long)b * 3 * N;

    for (int n = t; n < N; n += bd) dist[n] = 1e10f;
    if (t == 0) { s_last = 0; s_sel[0] = 0; }
    __syncthreads();

    for (int it = 1; it < S; ++it) {
        const int last = s_last;
        const float lx = X[last], ly = X[N + last], lz = X[2 * N + last];
        float bestv = -1.0f; int besti = 0x7fffffff;
        for (int n = t; n < N; n += bd) {
            float dx = X[n] - lx, dy = X[N + n] - ly, dz = X[2 * N + n] - lz;
            float d  = dx * dx + dy * dy + dz * dz;
            float dd = fminf(dist[n], d);
            dist[n] = dd;
            if (dd > bestv) { bestv = dd; besti = n; }
        }
        rv[t] = bestv; ri[t] = besti;
        __syncthreads();
        for (int offr = bd >> 1; offr > 0; offr >>= 1) {
            if (t < offr) {
                if (rv[t + offr] > rv[t] ||
                    (rv[t + offr] == rv[t] && ri[t + offr] < ri[t])) {
                    rv[t] = rv[t + offr]; ri[t] = ri[t + offr];
                }
            }
            __syncthreads();
        }
        if (t == 0) { s_last = ri[0]; s_sel[it] = ri[0]; }
        __syncthreads();
    }
    for (int e = t; e < S; e += bd) fidx[(long long)b * S + e] = s_sel[e];
    for (int e = t; e < 3 * S; e += bd) {
        int c = e / S, sI = e % S;
        new_xyz[(long long)b * 3 * S + e] = X[(long long)c * N + s_sel[sI]];
    }
}

// ---------------------------------------------------------------------------
// 16-NN: one thread per query, register insertion sort (ascending distance,
// matching top_k(-d) ordering).
// ---------------------------------------------------------------------------
__global__ void knn_kernel(const float* __restrict__ q, const float* __restrict__ r,
                           int* __restrict__ idx, int Bb, int Sq, int Nr)
{
    int i = blockIdx.x * blockDim.x + threadIdx.x;
    if (i >= Bb * Sq) return;
    int b = i / Sq, sI = i % Sq;
    const float* Q = q + (long long)b * 3 * Sq;
    const float* R = r + (long long)b * 3 * Nr;
    float qx = Q[sI], qy = Q[Sq + sI], qz = Q[2 * Sq + sI];

    float dbest[16]; int ibest[16];
#pragma unroll
    for (int j = 0; j < 16; ++j) { dbest[j] = 3.4e38f; ibest[j] = 0; }

    for (int n = 0; n < Nr; ++n) {
        float dx = qx - R[n], dy = qy - R[Nr + n], dz = qz - R[2 * Nr + n];
        float d = dx * dx + dy * dy + dz * dz;
        if (d < dbest[15]) {
            int j = 15;
            while (j > 0 && dbest[j - 1] > d) {
                dbest[j] = dbest[j - 1]; ibest[j] = ibest[j - 1]; --j;
            }
            dbest[j] = d; ibest[j] = n;
        }
    }
    int* O = idx + (long long)i * 16;
#pragma unroll
    for (int j = 0; j < 16; ++j) O[j] = ibest[j];
}

// ---------------------------------------------------------------------------
// Fused SA layer: group(xyz,pts,idx) -> conv1 -> relu -> conv2 -> max over 16.
// One block per (b, s). LDS: feat[Cin][16], h1[C1][16].
// ---------------------------------------------------------------------------
__global__ __launch_bounds__(128)
void sa_group_conv(const float* __restrict__ xyz, const float* __restrict__ pts, int Cp,
                   const int* __restrict__ idx, const float* __restrict__ new_xyz,
                   const float* __restrict__ w1, const float* __restrict__ b1, int C1,
                   const float* __restrict__ w2, const float* __restrict__ b2, int C2,
                   float* __restrict__ outp, int N, int S)
{
    extern __shared__ float sm[];
    const int Cin = 3 + Cp;
    float* feat = sm;
    float* h1   = sm + Cin * 16;
    __shared__ int sj[16];

    const int bs = blockIdx.x;
    const int b  = bs / S;
    const int sI = bs % S;
    const int t  = threadIdx.x;
    const int bd = blockDim.x;

    const int* I = idx + ((long long)b * S + sI) * 16;
    if (t < 16) sj[t] = I[t];
    __syncthreads();

    const float* X  = xyz + (long long)b * 3 * N;
    const float* P  = pts + (long long)b * Cp * N;
    const float* NX = new_xyz + (long long)b * 3 * S;

    for (int e = t; e < Cin * 16; e += bd) {
        int c = e >> 4, j = e & 15;
        int n = sj[j];
        feat[e] = (c < 3) ? (X[(long long)c * N + n] - NX[(long long)c * S + sI])
                          : P[(long long)(c - 3) * N + n];
    }
    __syncthreads();

    for (int e = t; e < C1 * 16; e += bd) {
        int o = e >> 4, j = e & 15;
        float acc = b1[o];
        const float* W = w1 + (long long)o * Cin;
        for (int c = 0; c < Cin; ++c) acc += W[c] * feat[(c << 4) + j];
        h1[e] = fmaxf(acc, 0.0f);
    }
    __syncthreads();

    for (int o2 = t; o2 < C2; o2 += bd) {
        const float* W = w2 + (long long)o2 * C1;
        float mx = -3.4e38f;
        for (int j = 0; j < 16; ++j) {
            float acc = b2[o2];
            for (int o = 0; o < C1; ++o) acc += W[o] * h1[(o << 4) + j];
            mx = fmaxf(mx, acc);
        }
        outp[((long long)b * C2 + o2) * S + sI] = mx;
    }
}

// ---------------------------------------------------------------------------
// Fused transformer position encoding:
//   pos_rel -> conv p1(64x3) -> BN -> relu -> conv p2(32x64) -> pe[B,32,S,16]
// One block per (b,s); pos_rel never materialized in memory.
// ---------------------------------------------------------------------------
__global__ __launch_bounds__(128)
void pe_kernel(const float* __restrict__ pos, const int* __restrict__ idx,
               const float* __restrict__ p1w, const float* __restrict__ p1b,
               const float* __restrict__ p1g, const float* __restrict__ p1be,
               const float* __restrict__ p2w, const float* __restrict__ p2b,
               float* __restrict__ pe, int S)
{
    __shared__ float pr[3][16];
    __shared__ float h[64 * 16];

    const int bs = blockIdx.x;
    const int b  = bs / S;
    const int sI = bs % S;
    const int t  = threadIdx.x;
    const float* P = pos + (long long)b * 3 * S;
    const int*   I = idx + ((long long)b * S + sI) * 16;

    if (t < 48) {
        int c = t / 16, j = t % 16;
        pr[c][j] = P[(long long)c * S + sI] - P[(long long)c * S + I[j]];
    }
    __syncthreads();

    for (int e = t; e < 64 * 16; e += blockDim.x) {
        int o = e >> 4, j = e & 15;
        float acc = p1b[o]
                  + p1w[o * 3 + 0] * pr[0][j]
                  + p1w[o * 3 + 1] * pr[1][j]
                  + p1w[o * 3 + 2] * pr[2][j];
        acc = acc * p1g[o] + p1be[o];
        h[e] = fmaxf(acc, 0.0f);
    }
    __syncthreads();

    for (int e = t; e < 32 * 16; e += blockDim.x) {
        int o = e >> 4, j = e & 15;
        float acc = p2b[o];
        const float* W = p2w + (long long)o * 64;
        for (int k = 0; k < 64; ++k) acc += W[k] * h[(k << 4) + j];
        pe[(((long long)b * 32 + o) * S + sI) * 16 + j] = acc;
    }
}

// ---------------------------------------------------------------------------
// t = (qu - group(key, idx)) + pe       layout (B, D, S, 16)
// ---------------------------------------------------------------------------
__global__ void build_t_kernel(const float* __restrict__ qu, const float* __restrict__ key,
                               const int* __restrict__ idx, const float* __restrict__ pe,
                               float* __restrict__ outp, int Bb, int D, int S)
{
    long long i = (long long)blockIdx.x * blockDim.x + threadIdx.x;
    long long total = (long long)Bb * D * S * 16;
    if (i >= total) return;
    int j = (int)(i & 15);
    long long r = i >> 4;
    int sI = (int)(r % S);
    long long r2 = r / S;
    int c = (int)(r2 % D);
    int b = (int)(r2 / D);
    int nn = idx[((long long)b * S + sI) * 16 + j];
    outp[i] = qu[((long long)b * D + c) * S + sI]
            - key[((long long)b * D + c) * S + nn] + pe[i];
}

// ---------------------------------------------------------------------------
// Fused attention tail: a1(128x32)+BN+relu -> a2(32x128) -> softmax over 16
// -> agg = sum_j a * (val + pe).  One block per (b,s); the huge (B,128,S,16)
// activation only ever lives in LDS.
// ---------------------------------------------------------------------------
__global__ __launch_bounds__(128)
void attn_kernel(const float* __restrict__ tb, const float* __restrict__ val,
                 const float* __restrict__ pe,
                 const float* __restrict__ a1w, const float* __restrict__ a1b,
                 const float* __restrict__ a1g, const float* __restrict__ a1be,
                 const float* __restrict__ a2w, const float* __restrict__ a2b,
                 float* __restrict__ agg, int S)
{
    __shared__ float ts[32 * 16];
    __shared__ float h[128 * 16];
    __shared__ float av[32 * 16];

    const int bs = blockIdx.x;
    const int b  = bs / S;
    const int sI = bs % S;
    const int t  = threadIdx.x;

    for (int e = t; e < 32 * 16; e += blockDim.x) {
        int o = e >> 4, j = e & 15;
        ts[e] = tb[(((long long)b * 32 + o) * S + sI) * 16 + j];
    }
    __syncthreads();

    for (int e = t; e < 128 * 16; e += blockDim.x) {
        int o = e >> 4, j = e & 15;
        float acc = a1b[o];
        const float* W = a1w + (long long)o * 32;
        for (int k = 0; k < 32; ++k) acc += W[k] * ts[(k << 4) + j];
        acc = acc * a1g[o] + a1be[o];
        h[e] = fmaxf(acc, 0.0f);
    }
    __syncthreads();

    for (int e = t; e < 32 * 16; e += blockDim.x) {
        int o = e >> 4, j = e & 15;
        float acc = a2b[o];
        const float* W = a2w + (long long)o * 128;
        for (int k = 0; k < 128; ++k) acc += W[k] * h[(k << 4) + j];
        av[e] = acc;
    }
    __syncthreads();

    if (t < 32) {
        float* row = av + (t << 4);
        float m = row[0];
#pragma unroll
        for (int j = 1; j < 16; ++j) m = fmaxf(m, row[j]);
        float ex[16]; float ssum = 0.0f;
#pragma unroll
        for (int j = 0; j < 16; ++j) { ex[j] = expf(row[j] - m); ssum += ex[j]; }
        float inv = 1.0f / ssum;
        float v = val[((long long)b * 32 + t) * S + sI];
        const float* PE = pe + (((long long)b * 32 + t) * S + sI) * 16;
        float s = 0.0f;
#pragma unroll
        for (int j = 0; j < 16; ++j) s += ex[j] * inv * (v + PE[j]);
        agg[((long long)b * 32 + t) * S + sI] = s;
    }
}

// ---------------------------------------------------------------------------
// concat along channels: out(B, Ca+Cb, S)
// ---------------------------------------------------------------------------
__global__ void concat_ch(const float* __restrict__ a, int Ca,
                          const float* __restrict__ bb, int Cb,
                          float* __restrict__ o, int Bb, int S)
{
    int i = blockIdx.x * blockDim.x + threadIdx.x;
    int Ct = Ca + Cb;
    int total = Bb * Ct * S;
    if (i >= total) return;
    int sI = i % S;
    int c  = (i / S) % Ct;
    int b  = i / (S * Ct);
    o[i] = (c < Ca) ? a[((long long)b * Ca + c) * S + sI]
                    : bb[((long long)b * Cb + (c - Ca)) * S + sI];
}

__global__ void max_last(const float* __restrict__ h, float* __restrict__ o,
                         long long rows, int S)
{
    long long i = (long long)blockIdx.x * blockDim.x + threadIdx.x;
    if (i >= rows) return;
    const float* p = h + i * S;
    float m = p[0];
    for (int j = 1; j < S; ++j) m = fmaxf(m, p[j]);
    o[i] = m;
}

// fill the broadcast-x part (cols 8..519) of a decoder concat buffer (F,B,520)
__global__ void fill_bcast(float* __restrict__ feat, const float* __restrict__ x,
                           int F, int Bb)
{
    int i = blockIdx.x * blockDim.x + threadIdx.x;
    int total = F * Bb * 512;
    if (i >= total) return;
    int c = i % 512;
    int b = (i / 512) % Bb;
    int f = i / (512 * Bb);
    feat[((long long)f * Bb + b) * 520 + 8 + c] = x[(long long)b * 512 + c];
}

// ---------------------------------------------------------------------------
// Decoder stage 3 (K=64, N=8 or 3): scalar dot + scatter.
// mode 0: write into next-level concat buffer at row f' = f*KH + head, ld=520
// mode 1: final output, out[b, f', n]  (the reference's transpose(1,0,2))
// ---------------------------------------------------------------------------
__global__ void dec_stage3(const float* __restrict__ h2, const float* __restrict__ w3,
                           const float* __restrict__ b3, float* __restrict__ outp,
                           int M, int Bb, int cout, int head, int KH,
                           int mode, int doRelu, int ldout, int Ftot)
{
    int i = blockIdx.x * blockDim.x + threadIdx.x;
    int total = M * cout;
    if (i >= total) return;
    int m = i / cout, n = i % cout;
    const float* H = h2 + (long long)m * 64;
    const float* W = w3 + (long long)n * 64;
    float acc = b3[n];
#pragma unroll 8
    for (int k = 0; k < 64; ++k) acc += H[k] * W[k];
    if (doRelu) acc = fmaxf(acc, 0.0f);
    int f = m / Bb, b = m % Bb;
    int fp = f * KH + head;
    if (mode == 0) outp[((long long)fp * Bb + b) * ldout + n] = acc;
    else           outp[((long long)b * Ftot + fp) * cout + n] = acc;
}

// ===========================================================================
// Host-side helpers
// ===========================================================================
static inline void launch_gemm(hipStream_t st,
    const float* A, int lda, long long sA,
    const float* Bm, int ldb, long long sB, int transB,
    float* C, int ldc, long long sC,
    const float* bias, long long sBias, int biasPerCol,
    const float* gamma, const float* beta,
    const float* resid, int ldres, long long sRes,
    int M, int N, int K, int relu, int batch)
{
    dim3 grid((N + 15) / 16, (M + 15) / 16, batch);
    gemm_wmma_f32<<<grid, dim3(32, 1, 1), 0, st>>>(
        A, lda, sA, Bm, ldb, sB, transB, C, ldc, sC,
        bias, sBias, biasPerCol, gamma, beta, resid, ldres, sRes,
        M, N, K, relu);
}

struct TScratch { float *x1, *key, *qu, *val, *pe, *tb, *agg; int* knn; };

static void run_transformer(hipStream_t st, void* const* din, int base,
                            const float* x, const float* pos, float* xout,
                            int Bb, int cin, int n, TScratch& S)
{
    const int D = 32;
    const float* ls_w  = (const float*)din[base + 0];
    const float* ls_b  = (const float*)din[base + 1];
    const float* k_w   = (const float*)din[base + 2];
    const float* k_b   = (const float*)din[base + 3];
    const float* q_w   = (const float*)din[base + 4];
    const float* q_b   = (const float*)din[base + 5];
    const float* v_w   = (const float*)din[base + 6];
    const float* v_b   = (const float*)din[base + 7];
    const float* p1_w  = (const float*)din[base + 8];
    const float* p1_b  = (const float*)din[base + 9];
    const float* p1_g  = (const float*)din[base + 10];
    const float* p1_be = (const float*)din[base + 11];
    const float* p2_w  = (const float*)din[base + 12];
    const float* p2_b  = (const float*)din[base + 13];
    const float* a1_w  = (const float*)din[base + 14];
    const float* a1_b  = (const float*)din[base + 15];
    const float* a1_g  = (const float*)din[base + 16];
    const float* a1_be = (const float*)din[base + 17];
    const float* a2_w  = (const float*)din[base + 18];
    const float* a2_b  = (const float*)din[base + 19];
    const float* le_w  = (const float*)din[base + 20];
    const float* le_b  = (const float*)din[base + 21];

    // self-KNN over positions
    { int tot = Bb * n;
      knn_kernel<<<(tot + 63) / 64, 64, 0, st>>>(pos, pos, S.knn, Bb, n, n); }

    // x1 = ls(x); key/qu/val = conv32x32(x1)   [WMMA, per-batch]
    launch_gemm(st, ls_w, cin, 0, x, n, (long long)cin * n, 0, S.x1, n, (long long)D * n,
                ls_b, 0, 0, nullptr, nullptr, nullptr, 0, 0, D, n, cin, 0, Bb);
    launch_gemm(st, k_w, D, 0, S.x1, n, (long long)D * n, 0, S.key, n, (long long)D * n,
                k_b, 0, 0, nullptr, nullptr, nullptr, 0, 0, D, n, D, 0, Bb);
    launch_gemm(st, q_w, D, 0, S.x1, n, (long long)D * n, 0, S.qu, n, (long long)D * n,
                q_b, 0, 0, nullptr, nullptr, nullptr, 0, 0, D, n, D, 0, Bb);
    launch_gemm(st, v_w, D, 0, S.x1, n, (long long)D * n, 0, S.val, n, (long long)D * n,
                v_b, 0, 0, nullptr, nullptr, nullptr, 0, 0, D, n, D, 0, Bb);

    // fused position-encoding MLP
    pe_kernel<<<Bb * n, 128, 0, st>>>(pos, S.knn, p1_w, p1_b, p1_g, p1_be,
                                      p2_w, p2_b, S.pe, n);
    // t = qk_rel + pe
    { long long tot = (long long)Bb * D * n * 16;
      build_t_kernel<<<(int)((tot + 255) / 256), 256, 0, st>>>(
          S.qu, S.key, S.knn, S.pe, S.tb, Bb, D, n); }
    // fused a1/a2/softmax/aggregate
    attn_kernel<<<Bb * n, 128, 0, st>>>(S.tb, S.val, S.pe,
                                        a1_w, a1_b, a1_g, a1_be, a2_w, a2_b,
                                        S.agg, n);
    // out = le(agg) + residual x
    launch_gemm(st, le_w, D, 0, S.agg, n, (long long)D * n, 0, xout, n, (long long)cin * n,
                le_b, 0, 0, nullptr, nullptr, x, n, (long long)cin * n,
                cin, n, D, 0, Bb);
}

static void dec_run(hipStream_t st, const float* feat, int F, int Bb, int C,
                    void* const* din, int pbase, int KH, int cout,
                    float* h1, float* h2, float* outbuf, int mode, int relu3, int Ftot)
{
    const float* w1 = (const float*)din[pbase + 0];
    const float* b1 = (const float*)din[pbase + 1];
    const float* w2 = (const float*)din[pbase + 2];
    const float* b2 = (const float*)din[pbase + 3];
    const float* w3 = (const float*)din[pbase + 4];
    const float* b3 = (const float*)din[pbase + 5];
    const int M = F * Bb;
    for (int k = 0; k < KH; ++k) {
        // stage 1: (M x C) * w1[k]^T -> (M x 256), bias per col, relu
        launch_gemm(st, feat, C, 0, w1 + (long long)k * 256 * C, C, 0, 1,
                    h1, 256, 0, b1 + (long long)k * 256, 0, 1,
                    nullptr, nullptr, nullptr, 0, 0, M, 256, C, 1, 1);
        // stage 2: (M x 256) * w2[k]^T -> (M x 64), bias per col, relu
        launch_gemm(st, h1, 256, 0, w2 + (long long)k * 64 * 256, 256, 0, 1,
                    h2, 64, 0, b2 + (long long)k * 64, 0, 1,
                    nullptr, nullptr, nullptr, 0, 0, M, 64, 256, 1, 1);
        // stage 3: scalar dot + scatter into concat buffer / final output
        int tot = M * cout;
        dec_stage3<<<(tot + 255) / 256, 256, 0, st>>>(
            h2, w3 + (long long)k * cout * 64, b3 + (long long)k * cout,
            outbuf, M, Bb, cout, k, KH, mode, relu3, 520, Ftot);
    }
}

// ===========================================================================
// kernel_launch — full forward pass
// Param order follows setup_inputs() dict insertion order.
// Workspace use: ~43 MB, bump-allocated from d_ws.
// ===========================================================================
extern "C" void kernel_launch(void* const* d_in, const int* in_sizes, int n_in,
                              void* d_out, int out_size, void* d_ws, size_t ws_size,
                              hipStream_t stream)
{
    (void)in_sizes; (void)n_in; (void)out_size; (void)ws_size;
    const int Bb = 16, N = 2048, S1 = 256, S2 = 64;
    const float* pcd = (const float*)d_in[0];

    char* wsb = (char*)d_ws;
    size_t off = 0;
    auto alloc_f = [&](size_t elems) -> float* {
        float* p = (float*)(wsb + off);
        off += ((elems * sizeof(float)) + 255) & ~(size_t)255;
        return p;
    };
    auto alloc_i = [&](size_t elems) -> int* {
        int* p = (int*)(wsb + off);
        off += ((elems * sizeof(int)) + 255) & ~(size_t)255;
        return p;
    };

    float* xyz     = alloc_f((size_t)Bb * 3 * N);
    float* l1_xyz  = alloc_f((size_t)Bb * 3 * S1);
    float* l1_ptsA = alloc_f((size_t)Bb * 64 * S1);
    float* l1_ptsB = alloc_f((size_t)Bb * 64 * S1);
    float* l2_xyz  = alloc_f((size_t)Bb * 3 * S2);
    float* l2_ptsA = alloc_f((size_t)Bb * 128 * S2);
    float* l2_ptsB = alloc_f((size_t)Bb * 128 * S2);
    float* xg      = alloc_f((size_t)Bb * 512);
    int*   fidx    = alloc_i((size_t)Bb * S1);
    int*   knn     = alloc_i((size_t)Bb * S1 * 16);

    TScratch ts;
    ts.x1  = alloc_f((size_t)Bb * 32 * S1);
    ts.key = alloc_f((size_t)Bb * 32 * S1);
    ts.qu  = alloc_f((size_t)Bb * 32 * S1);
    ts.val = alloc_f((size_t)Bb * 32 * S1);
    ts.pe  = alloc_f((size_t)Bb * 32 * S1 * 16);
    ts.tb  = alloc_f((size_t)Bb * 32 * S1 * 16);
    ts.agg = alloc_f((size_t)Bb * 32 * S1);
    ts.knn = knn;

    float* feat131 = alloc_f((size_t)Bb * 131 * S2);
    float* h256    = alloc_f((size_t)Bb * 256 * S2);
    float* h512    = alloc_f((size_t)Bb * 512 * S2);

    float* featS1 = alloc_f((size_t)4   * Bb * 520);
    float* featS2 = alloc_f((size_t)32  * Bb * 520);
    float* featL  = alloc_f((size_t)256 * Bb * 520);
    float* dh1    = alloc_f((size_t)4096 * 256);
    float* dh2    = alloc_f((size_t)4096 * 64);

    // ---- input transpose ----
    { int tot = Bb * N * 3;
      transpose_in<<<(tot + 255) / 256, 256, 0, stream>>>(pcd, xyz, Bb, N); }

    // ---- SA1: fps(2048->256), knn16, fused group+MLP+max ----
    fps_kernel<<<Bb, 256, 0, stream>>>(xyz, fidx, l1_xyz, N, S1);
    { int tot = Bb * S1;
      knn_kernel<<<(tot + 63) / 64, 64, 0, stream>>>(l1_xyz, xyz, knn, Bb, S1, N); }
    {
        size_t shm = (size_t)((6 + 32) * 16) * sizeof(float);
        sa_group_conv<<<Bb * S1, 128, shm, stream>>>(
            xyz, xyz, 3, knn, l1_xyz,
            (const float*)d_in[1], (const float*)d_in[2], 32,
            (const float*)d_in[3], (const float*)d_in[4], 64,
            l1_ptsA, N, S1);
    }

    // ---- transformer 1 (cin=64, n=256) ----
    run_transformer(stream, d_in, 13, l1_ptsA, l1_xyz, l1_ptsB, Bb, 64, S1, ts);

    // ---- SA2: fps(256->64), knn16, fused group+MLP+max ----
    fps_kernel<<<Bb, 256, 0, stream>>>(l1_xyz, fidx, l2_xyz, S1, S2);
    { int tot = Bb * S2;
      knn_kernel<<<(tot + 63) / 64, 64, 0, stream>>>(l2_xyz, l1_xyz, knn, Bb, S2, S1); }
    {
        size_t shm = (size_t)((67 + 64) * 16) * sizeof(float);
        sa_group_conv<<<Bb * S2, 128, shm, stream>>>(
            l1_xyz, l1_ptsB, 64, knn, l2_xyz,
            (const float*)d_in[5], (const float*)d_in[6], 64,
            (const float*)d_in[7], (const float*)d_in[8], 128,
            l2_ptsA, S1, S2);
    }

    // ---- transformer 2 (cin=128, n=64) ----
    run_transformer(stream, d_in, 35, l2_ptsA, l2_xyz, l2_ptsB, Bb, 128, S2, ts);

    // ---- SA-all: concat -> conv(256x131)+relu -> conv(512x256) -> max ----
    { int tot = Bb * 131 * S2;
      concat_ch<<<(tot + 255) / 256, 256, 0, stream>>>(l2_xyz, 3, l2_ptsB, 128,
                                                       feat131, Bb, S2); }
    launch_gemm(stream, (const float*)d_in[9], 131, 0,
                feat131, S2, (long long)131 * S2, 0,
                h256, S2, (long long)256 * S2,
                (const float*)d_in[10], 0, 0, nullptr, nullptr, nullptr, 0, 0,
                256, S2, 131, 1, Bb);
    launch_gemm(stream, (const float*)d_in[11], 256, 0,
                h256, S2, (long long)256 * S2, 0,
                h512, S2, (long long)512 * S2,
                (const float*)d_in[12], 0, 0, nullptr, nullptr, nullptr, 0, 0,
                512, S2, 256, 0, Bb);
    { long long rows = (long long)Bb * 512;
      max_last<<<(int)((rows + 255) / 256), 256, 0, stream>>>(h512, xg, rows, S2); }

    // ---- decoder: d1(K=4) -> d2 -> d3 -> d4 (per-head WMMA GEMM chains) ----
    { int tot = 4 * Bb * 512;
      fill_bcast<<<(tot + 255) / 256, 256, 0, stream>>>(featS1, xg, 4, Bb); }
    dec_run(stream, xg, 1, Bb, 512, d_in, 57, 4, 8, dh1, dh2, featS1, 0, 1, 0);

    { int tot = 32 * Bb * 512;
      fill_bcast<<<(tot + 255) / 256, 256, 0, stream>>>(featS2, xg, 32, Bb); }
    dec_run(stream, featS1, 4, Bb, 520, d_in, 63, 8, 8, dh1, dh2, featS2, 0, 1, 0);

    { int tot = 256 * Bb * 512;
      fill_bcast<<<(tot + 255) / 256, 256, 0, stream>>>(featL, xg, 256, Bb); }
    dec_run(stream, featS2, 32, Bb, 520, d_in, 69, 8, 8, dh1, dh2, featL, 0, 1, 0);

    // d4 writes the final (16, 2048, 3) output directly (transposed store).
    dec_run(stream, featL, 256, Bb, 520, d_in, 75, 8, 3, dh1, dh2,
            (float*)d_out, 1, 0, 2048);
}